// sur_GNN_63574105915529
// MI455X (gfx1250) — compile-verified
//
#include <hip/hip_runtime.h>
#include <hip/hip_bf16.h>

typedef __attribute__((ext_vector_type(2))) float v2f;
typedef __attribute__((ext_vector_type(8))) float v8f;

#define N_NODES 8192
#define FDIM    256
#define HDIM    128
#define QDIM    64
#define KT      32
#define LDA_S   33   // padded LDS stride for A tile (avoid bank conflicts)

// ---------------------------------------------------------------------------
// Generic WMMA f32 GEMM:  D[M, 128*G] = A[M,K] @ B[K, 128*G]  (row-major)
// Optional epilogue: v = rowscale[row]*v + bias[col]; relu.
// Block = 256 threads (8 waves). Each block covers 32 rows x (128*G) cols:
//   wave w -> 16-col strip w, 2 row-tiles, G column groups.
// A tile (32xKT) and B tile (KT x 128G) staged in LDS per K-chunk.
// ---------------------------------------------------------------------------
template <int G>
__global__ __launch_bounds__(256) void gemm_wmma(
    const float* __restrict__ A, const float* __restrict__ B,
    float* __restrict__ D, int lda, int ldb, int ldd, int K,
    const float* __restrict__ rowscale, const float* __restrict__ bias,
    int relu) {
  constexpr int CT = 128 * G;
  constexpr int LDB_S = CT + 4;  // padded stride
  __shared__ float As[32 * LDA_S];
  __shared__ float Bs[KT * LDB_S];

  const int tid  = threadIdx.x;
  const int wave = tid >> 5;
  const int lane = tid & 31;
  const int r16  = lane & 15;
  const int hi   = lane >> 4;   // 0 or 1 (half-wave)
  const int koff = hi * 2;      // f32 WMMA A/B frag K offset for lanes 16-31
  const int r0   = blockIdx.x * 32;
  const int lcol = wave * 16 + r16;  // 0..127 within a column group

  v8f acc[G][2];
#pragma unroll
  for (int g = 0; g < G; ++g)
#pragma unroll
    for (int t = 0; t < 2; ++t) acc[g][t] = (v8f)0.0f;

  for (int kb = 0; kb < K; kb += KT) {
    __syncthreads();
    // Stage A: 32 rows x KT (coalesced 32-lane rows)
    for (int idx = tid; idx < 32 * KT; idx += 256) {
      int r = idx >> 5, k = idx & (KT - 1);
      As[r * LDA_S + k] = A[(size_t)(r0 + r) * lda + (kb + k)];
    }
    // Stage B: KT rows x CT cols
    for (int idx = tid; idx < KT * CT; idx += 256) {
      int k = idx / CT, c = idx - k * CT;
      Bs[k * LDB_S + c] = B[(size_t)(kb + k) * ldb + c];
    }
    __syncthreads();

#pragma unroll
    for (int kk = 0; kk < KT; kk += 4) {
      v2f a[2];
#pragma unroll
      for (int t = 0; t < 2; ++t) {
        a[t].x = As[(t * 16 + r16) * LDA_S + kk + koff];
        a[t].y = As[(t * 16 + r16) * LDA_S + kk + koff + 1];
      }
#pragma unroll
      for (int g = 0; g < G; ++g) {
        v2f b;
        b.x = Bs[(kk + koff) * LDB_S + g * 128 + lcol];
        b.y = Bs[(kk + koff + 1) * LDB_S + g * 128 + lcol];
#pragma unroll
        for (int t = 0; t < 2; ++t) {
          acc[g][t] = __builtin_amdgcn_wmma_f32_16x16x4_f32(
              false, a[t], false, b, (short)0, acc[g][t], false, false);
        }
      }
    }
  }

  // Epilogue + store (C/D layout: VGPR r -> row = t*16 + hi*8 + r, col = lcol)
#pragma unroll
  for (int g = 0; g < G; ++g) {
    const int gc = g * 128 + lcol;
    const float bc = bias ? bias[gc] : 0.0f;
#pragma unroll
    for (int t = 0; t < 2; ++t) {
#pragma unroll
      for (int r = 0; r < 8; ++r) {
        int gr = r0 + t * 16 + hi * 8 + r;
        float v = acc[g][t][r];
        if (rowscale) v *= rowscale[gr];
        v += bc;
        if (relu) v = fmaxf(v, 0.0f);
        D[(size_t)gr * ldd + gc] = v;
      }
    }
  }
}

// ---------------------------------------------------------------------------
// Small helper kernels
// ---------------------------------------------------------------------------
__global__ void rowsum_kernel(const float* __restrict__ adj, float* __restrict__ deg) {
  __shared__ float red[256];
  int i = blockIdx.x;
  float s = 0.0f;
  for (int j = threadIdx.x; j < N_NODES; j += 256)
    s += adj[(size_t)i * N_NODES + j];
  red[threadIdx.x] = s;
  __syncthreads();
  for (int off = 128; off > 0; off >>= 1) {
    if ((int)threadIdx.x < off) red[threadIdx.x] += red[threadIdx.x + off];
    __syncthreads();
  }
  if (threadIdx.x == 0) deg[i] = red[0];
}

__global__ void dvec_kernel(const float* __restrict__ deg, float* __restrict__ d) {
  int i = blockIdx.x * blockDim.x + threadIdx.x;
  if (i < N_NODES) {
    float v = deg[i];
    d[i] = (v > 0.0f) ? rsqrtf(v) : 0.0f;  // deg==0 -> inf -> 0 per reference
  }
}

// q may contain duplicates; indicator semantics require collapsing them.
__global__ void dedup_kernel(const int* __restrict__ q, int* __restrict__ keep) {
  int k = threadIdx.x;  // blockDim == QDIM
  int qi = q[k];
  int kp = 1;
  for (int j = 0; j < k; ++j)
    if (q[j] == qi) kp = 0;
  keep[k] = kp;
}

// s[i] = d[i] * sum_{unique j in q} adj[i,j] * d[j]
__global__ void svec_kernel(const float* __restrict__ adj, const int* __restrict__ q,
                            const int* __restrict__ keep, const float* __restrict__ d,
                            float* __restrict__ s) {
  __shared__ int   qs[QDIM];
  __shared__ float dq[QDIM];
  __shared__ int   ks[QDIM];
  if (threadIdx.x < QDIM) {
    int qi = q[threadIdx.x];
    qs[threadIdx.x] = qi;
    dq[threadIdx.x] = d[qi];
    ks[threadIdx.x] = keep[threadIdx.x];
  }
  __syncthreads();
  int i = blockIdx.x * blockDim.x + threadIdx.x;
  if (i < N_NODES) {
    float acc = 0.0f;
    for (int k = 0; k < QDIM; ++k)
      if (ks[k]) acc += adj[(size_t)i * N_NODES + qs[k]] * dq[k];
    s[i] = d[i] * acc;
  }
}

__global__ void bias_add_kernel(const float* a, const float* b, float* out) {
  int c = threadIdx.x;  // blockDim == HDIM
  out[c] = a[c] + b[c];
}

// hf_in[q[k], :] += Wlq  (unique q only)
__global__ void addq_kernel(float* __restrict__ hfin, const int* __restrict__ q,
                            const int* __restrict__ keep, const float* __restrict__ Wlq) {
  int k = blockIdx.x;  // QDIM blocks x HDIM threads
  if (keep[k]) {
    int i = q[k];
    hfin[(size_t)i * HDIM + threadIdx.x] += Wlq[threadIdx.x];
  }
}

// Layer 0 epilogue: Y = adj @ [Xh | Xf] (ld 256)
__global__ void layer0_post_kernel(const float* __restrict__ Y, const float* __restrict__ d,
                                   const float* __restrict__ s, const float* __restrict__ Wq0,
                                   const float* __restrict__ bq0, const float* __restrict__ b0,
                                   const float* __restrict__ bf0, float* __restrict__ hq,
                                   float* __restrict__ h, float* __restrict__ hf) {
  int idx = blockIdx.x * blockDim.x + threadIdx.x;
  int i = idx >> 7, c = idx & 127;
  if (i < N_NODES) {
    float di = d[i];
    float hqv = fmaxf(s[i] * Wq0[c] + bq0[c], 0.0f);
    float hv  = fmaxf(di * Y[(size_t)i * 256 + c] + b0[c], 0.0f);
    float hfv = fmaxf(hqv + hv + di * Y[(size_t)i * 256 + 128 + c] + bf0[c], 0.0f);
    hq[idx] = hqv; h[idx] = hv; hf[idx] = hfv;
  }
}

// Layers 1/2 epilogue: Y = adj @ [Xq | Xh | Xf] (ld 384)
__global__ void layer_post_kernel(const float* __restrict__ Y, const float* __restrict__ d,
                                  const float* __restrict__ bq, const float* __restrict__ bh,
                                  const float* __restrict__ bf, float* __restrict__ hq,
                                  float* __restrict__ h, float* __restrict__ hf, int relu) {
  int idx = blockIdx.x * blockDim.x + threadIdx.x;
  int i = idx >> 7, c = idx & 127;
  if (i < N_NODES) {
    float di = d[i];
    float hqv = di * Y[(size_t)i * 384 + c] + bq[c];
    float hv  = di * Y[(size_t)i * 384 + 128 + c] + bh[c];
    if (relu) { hqv = fmaxf(hqv, 0.0f); hv = fmaxf(hv, 0.0f); }
    float hfv = hqv + hv + di * Y[(size_t)i * 384 + 256 + c] + bf[c];
    if (relu) hfv = fmaxf(hfv, 0.0f);
    hq[idx] = hqv; h[idx] = hv; hf[idx] = hfv;
  }
}

__global__ void norm_kernel(const float* __restrict__ h_, float* __restrict__ nrm) {
  int i = blockIdx.x * blockDim.x + threadIdx.x;
  if (i < N_NODES) {
    float s = 0.0f;
    for (int c = 0; c < HDIM; ++c) {
      float v = h_[(size_t)i * HDIM + c];
      s += v * v;
    }
    nrm[i] = sqrtf(s);
  }
}

__global__ void sim_kernel(const float* __restrict__ h_, const float* __restrict__ nrm,
                           const int* __restrict__ q, float* __restrict__ out) {
  __shared__ float qrow[HDIM];
  __shared__ float qn;
  int k = blockIdx.y;
  if ((int)threadIdx.x < HDIM) qrow[threadIdx.x] = h_[(size_t)q[k] * HDIM + threadIdx.x];
  if (threadIdx.x == 0) qn = nrm[q[k]];
  __syncthreads();
  int i = blockIdx.x * blockDim.x + threadIdx.x;
  if (i < N_NODES) {
    float dot = 0.0f;
    for (int c = 0; c < HDIM; ++c) dot += qrow[c] * h_[(size_t)i * HDIM + c];
    float sim = dot / (qn * nrm[i]);
    out[(size_t)k * N_NODES + i] = 1.0f / (1.0f + expf(-sim));
  }
}

// ---------------------------------------------------------------------------
extern "C" void kernel_launch(void* const* d_in, const int* in_sizes, int n_in,
                              void* d_out, int out_size, void* d_ws, size_t ws_size,
                              hipStream_t stream) {
  (void)in_sizes; (void)n_in; (void)out_size; (void)ws_size;
  const int*   q    = (const int*)d_in[0];
  const float* feats= (const float*)d_in[1];
  const float* adj  = (const float*)d_in[2];
  const float* Wq0  = (const float*)d_in[3];  const float* bq0 = (const float*)d_in[4];
  const float* Wq1  = (const float*)d_in[5];  const float* bq1 = (const float*)d_in[6];
  const float* Wq2  = (const float*)d_in[7];  const float* bq2 = (const float*)d_in[8];
  const float* W0   = (const float*)d_in[9];  const float* b0  = (const float*)d_in[10];
  const float* W1   = (const float*)d_in[11]; const float* b1  = (const float*)d_in[12];
  const float* W2   = (const float*)d_in[13]; const float* b2  = (const float*)d_in[14];
  const float* Wf0  = (const float*)d_in[15]; const float* bf0 = (const float*)d_in[16];
  const float* Wf1  = (const float*)d_in[17]; const float* bf1 = (const float*)d_in[18];
  const float* Wf2  = (const float*)d_in[19]; const float* bf2 = (const float*)d_in[20];
  const float* Wlq  = (const float*)d_in[21]; const float* blq = (const float*)d_in[22];
  const float* Wlf  = (const float*)d_in[23]; const float* blf = (const float*)d_in[24];
  const float* M1w  = (const float*)d_in[25]; const float* M1b = (const float*)d_in[26];
  const float* M2w  = (const float*)d_in[27]; const float* M2b = (const float*)d_in[28];
  float* out = (float*)d_out;

  // Workspace layout (floats)
  float* w = (float*)d_ws;
  size_t o = 0;
  float* dvec = w + o; o += N_NODES;
  float* deg  = w + o; o += N_NODES;   // reused as row-norms at the end
  float* svec = w + o; o += N_NODES;
  float* btmp = w + o; o += HDIM;
  int*   keep = (int*)(w + o); o += QDIM;
  float* hq   = w + o; o += (size_t)N_NODES * HDIM;  // reused as h_ at the end
  float* h    = w + o; o += (size_t)N_NODES * HDIM;
  float* hf   = w + o; o += (size_t)N_NODES * HDIM;
  float* hfin = w + o; o += (size_t)N_NODES * HDIM;  // reused as MLP temp
  float* Xs   = w + o; o += (size_t)N_NODES * 384;
  float* Y    = w + o; o += (size_t)N_NODES * 384;

  const int GB = N_NODES / 32;            // gemm row blocks
  const int EB = (N_NODES * HDIM) / 256;  // elementwise blocks

  // ---- degree / normalization vectors ----
  rowsum_kernel<<<N_NODES, 256, 0, stream>>>(adj, deg);
  dvec_kernel<<<N_NODES / 256, 256, 0, stream>>>(deg, dvec);
  dedup_kernel<<<1, QDIM, 0, stream>>>(q, keep);
  svec_kernel<<<N_NODES / 256, 256, 0, stream>>>(adj, q, keep, dvec, svec);

  // ---- layer 0 skinny operands ----
  // Xs[:,0:128] = d ∘ (feats @ W0)
  gemm_wmma<1><<<GB, 256, 0, stream>>>(feats, W0, Xs, FDIM, HDIM, 256, FDIM, dvec, nullptr, 0);
  // hfin = feats @ Wlf + (blf + blq), then += indicator ⊗ Wlq
  bias_add_kernel<<<1, HDIM, 0, stream>>>(blf, blq, btmp);
  gemm_wmma<1><<<GB, 256, 0, stream>>>(feats, Wlf, hfin, FDIM, HDIM, HDIM, FDIM, nullptr, btmp, 0);
  addq_kernel<<<QDIM, HDIM, 0, stream>>>(hfin, q, keep, Wlq);
  // Xs[:,128:256] = d ∘ (hfin @ Wf0)
  gemm_wmma<1><<<GB, 256, 0, stream>>>(hfin, Wf0, Xs + 128, HDIM, HDIM, 256, HDIM, dvec, nullptr, 0);
  // Y[:,0:256] = adj @ Xs  (single adjacency pass, 2 column groups)
  gemm_wmma<2><<<GB, 256, 0, stream>>>(adj, Xs, Y, N_NODES, 256, 256, N_NODES, nullptr, nullptr, 0);
  layer0_post_kernel<<<EB, 256, 0, stream>>>(Y, dvec, svec, Wq0, bq0, b0, bf0, hq, h, hf);

  // ---- layer 1 ----
  gemm_wmma<1><<<GB, 256, 0, stream>>>(hq, Wq1, Xs,       HDIM, HDIM, 384, HDIM, dvec, nullptr, 0);
  gemm_wmma<1><<<GB, 256, 0, stream>>>(h,  W1,  Xs + 128, HDIM, HDIM, 384, HDIM, dvec, nullptr, 0);
  gemm_wmma<1><<<GB, 256, 0, stream>>>(hf, Wf1, Xs + 256, HDIM, HDIM, 384, HDIM, dvec, nullptr, 0);
  gemm_wmma<3><<<GB, 256, 0, stream>>>(adj, Xs, Y, N_NODES, 384, 384, N_NODES, nullptr, nullptr, 0);
  layer_post_kernel<<<EB, 256, 0, stream>>>(Y, dvec, bq1, b1, bf1, hq, h, hf, 1);

  // ---- layer 2 (no relu) ----
  gemm_wmma<1><<<GB, 256, 0, stream>>>(hq, Wq2, Xs,       HDIM, HDIM, 384, HDIM, dvec, nullptr, 0);
  gemm_wmma<1><<<GB, 256, 0, stream>>>(h,  W2,  Xs + 128, HDIM, HDIM, 384, HDIM, dvec, nullptr, 0);
  gemm_wmma<1><<<GB, 256, 0, stream>>>(hf, Wf2, Xs + 256, HDIM, HDIM, 384, HDIM, dvec, nullptr, 0);
  gemm_wmma<3><<<GB, 256, 0, stream>>>(adj, Xs, Y, N_NODES, 384, 384, N_NODES, nullptr, nullptr, 0);
  layer_post_kernel<<<EB, 256, 0, stream>>>(Y, dvec, bq2, b2, bf2, hq, h, hf, 0);

  // ---- MLP head:  h_ = relu(hf @ M1w + M1b) @ M2w + M2b ----
  gemm_wmma<1><<<GB, 256, 0, stream>>>(hf,   M1w, hfin, HDIM, HDIM, HDIM, HDIM, nullptr, M1b, 1);
  gemm_wmma<1><<<GB, 256, 0, stream>>>(hfin, M2w, hq,   HDIM, HDIM, HDIM, HDIM, nullptr, M2b, 0);

  // ---- cosine similarity + sigmoid ----
  norm_kernel<<<N_NODES / 256, 256, 0, stream>>>(hq, deg);
  sim_kernel<<<dim3(N_NODES / 256, QDIM), 256, 0, stream>>>(hq, deg, q, out);
}